// Multiple_GIOU_loss_2_16071767622181
// MI455X (gfx1250) — compile-verified
//
#include <hip/hip_runtime.h>
#include <hip/hip_bf16.h>

#define TPB  256
#define GMAX 512

typedef int v4i __attribute__((ext_vector_type(4)));

__device__ __forceinline__ void copy16_g2l(const float4* gsrc, float4* ldst) {
#if __has_builtin(__builtin_amdgcn_global_load_async_to_lds_b128)
    __builtin_amdgcn_global_load_async_to_lds_b128(
        (v4i*)gsrc, (v4i*)ldst, /*offset=*/0, /*cpol=*/0);
#else
    *ldst = *gsrc;
#endif
}

__device__ __forceinline__ void wait_lds_copies() {
#if __has_builtin(__builtin_amdgcn_global_load_async_to_lds_b128)
  #if __has_builtin(__builtin_amdgcn_s_wait_asynccnt)
    __builtin_amdgcn_s_wait_asynccnt(0);
  #else
    asm volatile("s_wait_asynccnt 0" ::: "memory");
  #endif
#endif
}

// One block: 256 pred boxes of one batch vs all G target boxes (staged in LDS).
__global__ void __launch_bounds__(TPB)
giou_max_kernel(const float* __restrict__ tgt,   // imgs_box  [B,G,4]
                const float* __restrict__ pred,  // pre_BOX   [B,P,4]
                float* __restrict__ partial,     // [gridDim.y * gridDim.x]
                int P, int G) {
    __shared__ float4 sbox[GMAX];
    __shared__ float  sarea[GMAX];
    __shared__ float  sred[TPB / 32];

    const int b   = blockIdx.y;
    const int tid = threadIdx.x;

    // ---- stage target boxes for this batch into LDS (async on CDNA5) ----
    const float4* tgt4 = reinterpret_cast<const float4*>(tgt) + (size_t)b * G;
    for (int g = tid; g < G; g += TPB)
        copy16_g2l(tgt4 + g, &sbox[g]);
    wait_lds_copies();
    __syncthreads();

    // ---- precompute target areas once ----
    for (int g = tid; g < G; g += TPB) {
        float4 t  = sbox[g];
        float  tw = fmaxf(t.z - t.x + 1.0f, 0.0f);
        float  th = fmaxf(t.w - t.y + 1.0f, 0.0f);
        sarea[g]  = tw * th;
    }
    __syncthreads();

    // ---- per-thread: max GIoU over all targets ----
    const int p  = blockIdx.x * TPB + tid;
    float val = 0.0f;
    if (p < P) {
        float4 pb = reinterpret_cast<const float4*>(pred)[(size_t)b * P + p];
        float pw = fmaxf(pb.z - pb.x + 1.0f, 0.0f);
        float ph = fmaxf(pb.w - pb.y + 1.0f, 0.0f);
        float pa = pw * ph;

        float m = -2.0f;
        #pragma unroll 4
        for (int g = 0; g < G; ++g) {
            float4 t  = sbox[g];      // LDS broadcast (all lanes same addr)
            float  ta = sarea[g];

            float ix1 = fmaxf(pb.x, t.x), iy1 = fmaxf(pb.y, t.y);
            float ix2 = fminf(pb.z, t.z), iy2 = fminf(pb.w, t.w);
            float iw  = fmaxf(ix2 - ix1 + 1.0f, 0.0f);
            float ih  = fmaxf(iy2 - iy1 + 1.0f, 0.0f);
            float inter = iw * ih;
            float uni   = pa + ta - inter;
            float iou   = fmaxf(inter * __builtin_amdgcn_rcpf(uni), 1e-6f);

            float ox1 = fminf(pb.x, t.x), oy1 = fminf(pb.y, t.y);
            float ox2 = fmaxf(pb.z, t.z), oy2 = fmaxf(pb.w, t.w);
            float ow  = fmaxf(ox2 - ox1 + 1.0f, 0.0f);
            float oh  = fmaxf(oy2 - oy1 + 1.0f, 0.0f);
            float outer = ow * oh;

            float giou = iou - (outer - uni) * __builtin_amdgcn_rcpf(outer);
            giou = fminf(fmaxf(giou, -1.0f), 1.0f);
            m = fmaxf(m, giou);
        }
        val = 1.0f - m;
    }

    // ---- deterministic block reduction (wave32 shuffle tree) ----
    #pragma unroll
    for (int off = 16; off > 0; off >>= 1)
        val += __shfl_down(val, off, 32);
    if ((tid & 31) == 0) sred[tid >> 5] = val;
    __syncthreads();
    if (tid == 0) {
        float s = 0.0f;
        #pragma unroll
        for (int w = 0; w < TPB / 32; ++w) s += sred[w];
        partial[blockIdx.y * gridDim.x + blockIdx.x] = s;
    }
}

__global__ void __launch_bounds__(TPB)
final_reduce_kernel(const float* __restrict__ partial, float* __restrict__ out,
                    int n, float invB) {
    __shared__ float sred[TPB / 32];
    const int tid = threadIdx.x;
    float v = 0.0f;
    for (int i = tid; i < n; i += TPB) v += partial[i];  // fixed order per lane
    #pragma unroll
    for (int off = 16; off > 0; off >>= 1)
        v += __shfl_down(v, off, 32);
    if ((tid & 31) == 0) sred[tid >> 5] = v;
    __syncthreads();
    if (tid == 0) {
        float s = 0.0f;
        #pragma unroll
        for (int w = 0; w < TPB / 32; ++w) s += sred[w];
        out[0] = s * invB;
    }
}

extern "C" void kernel_launch(void* const* d_in, const int* in_sizes, int n_in,
                              void* d_out, int out_size, void* d_ws, size_t ws_size,
                              hipStream_t stream) {
    const float* imgs_box = (const float*)d_in[0];   // [B,G,4]
    const float* pre_BOX  = (const float*)d_in[1];   // [B,P,4]
    (void)n_in; (void)out_size; (void)ws_size;

    const int B = in_sizes[2];                  // labels: one per batch
    const int G = in_sizes[0] / (B * 4);        // 512 in reference
    const int P = in_sizes[1] / (B * 4);        // 2048 in reference

    float* partial = (float*)d_ws;
    const int bx = (P + TPB - 1) / TPB;

    dim3 grid(bx, B);
    giou_max_kernel<<<grid, TPB, 0, stream>>>(imgs_box, pre_BOX, partial, P, G);
    final_reduce_kernel<<<1, TPB, 0, stream>>>(partial, (float*)d_out,
                                               bx * B, 1.0f / (float)B);
}